// GAT_60181081752176
// MI455X (gfx1250) — compile-verified
//
#include <hip/hip_runtime.h>
#include <math.h>

// Problem constants (from reference)
#define NN    50000
#define NE    800000
#define ET    (NE + NN)      // edges + self loops = 850000
#define HID   128
#define NH    4
#define CH    32

typedef float v2f __attribute__((ext_vector_type(2)));
typedef float v8f __attribute__((ext_vector_type(8)));

// ---- monotone float<->int mapping for atomicMax-based segment max ----
__device__ __forceinline__ int f2ord(float f) {
    int i = __float_as_int(f);
    return (i >= 0) ? i : (i ^ 0x7fffffff);
}
__device__ __forceinline__ float ord2f(int i) {
    return __int_as_float((i >= 0) ? i : (i ^ 0x7fffffff));
}

// =====================================================================
// WMMA fp32 strip GEMM:  Y[M x (NT*16)] = X[M x K] @ W[K x (NT*16)] (+bias)
// - W is cooperatively staged in LDS once per workgroup (shared by 8 waves)
// - each wave computes a 16-row x NT*16-col strip: A-frag loaded once per
//   k-step (as one b64), reused across NT accumulators.
// M % 16 == 0 (50000 = 3125*16) -> no partial tiles; only workgroup/wave-
// uniform exits, so EXEC is all-ones at every WMMA.
// =====================================================================
template <int K, int NT>
__global__ void wmma_gemm_strip(const float* __restrict__ X,
                                const float* __restrict__ W,
                                const float* __restrict__ bias,
                                float* __restrict__ Y,
                                int nMtiles)
{
    constexpr int Nout = NT * 16;
    extern __shared__ float Wlds[];                  // K * Nout floats

    // cooperative stage of W into LDS (float4 granularity)
    for (int i = threadIdx.x * 4; i < K * Nout; i += blockDim.x * 4)
        *(float4*)&Wlds[i] = *(const float4*)&W[i];
    __syncthreads();

    int wave = (int)((blockIdx.x * blockDim.x + threadIdx.x) >> 5);
    if (wave >= nMtiles) return;                     // wave-uniform exit
    int lane = (int)(threadIdx.x & 31);

    int row = wave * 16 + (lane & 15);               // A row this lane supplies
    int col = lane & 15;                             // B col within a tile
    int kh  = (lane >> 4) << 1;                      // 0 or 2 (K sub-offset)

    v8f acc[NT];
#pragma unroll
    for (int t = 0; t < NT; ++t)
        acc[t] = (v8f){0.f, 0.f, 0.f, 0.f, 0.f, 0.f, 0.f, 0.f};

    const float* xrow = X + (size_t)row * K;
#pragma unroll
    for (int k = 0; k < K; k += 4) {
        v2f a = *(const v2f*)(xrow + k + kh);        // one global_load_b64
        const float* w0 = Wlds + (k + kh) * Nout + col;
#pragma unroll
        for (int t = 0; t < NT; ++t) {
            v2f b;
            b.x = w0[t * 16];                        // ds_load
            b.y = w0[t * 16 + Nout];
            acc[t] = __builtin_amdgcn_wmma_f32_16x16x4_f32(
                         false, a, false, b, (short)0, acc[t], false, false);
        }
    }

    int orow = wave * 16 + ((lane >> 4) << 3);
#pragma unroll
    for (int t = 0; t < NT; ++t) {
        int ocol = t * 16 + (lane & 15);
        float badd = bias ? bias[ocol] : 0.f;
#pragma unroll
        for (int r = 0; r < 8; ++r)
            Y[(size_t)(orow + r) * Nout + ocol] = acc[t][r] + badd;
    }
}

// ---- per-(node,head) attention logit dot products -------------------
__global__ void alpha_kernel(const float* __restrict__ Hm,
                             const float* __restrict__ a_s,
                             const float* __restrict__ a_d,
                             float* __restrict__ as_out,
                             float* __restrict__ ad_out)
{
    int tid = blockIdx.x * blockDim.x + threadIdx.x;
    if (tid >= NN * NH) return;
    int n = tid >> 2, h = tid & 3;
    const float* hp = Hm  + n * HID + h * CH;
    const float* sp = a_s + h * CH;
    const float* dp = a_d + h * CH;
    float s = 0.f, d = 0.f;
#pragma unroll
    for (int c = 0; c < CH; c += 4) {
        float4 hv = *(const float4*)(hp + c);
        float4 sv = *(const float4*)(sp + c);
        float4 dv = *(const float4*)(dp + c);
        s += hv.x * sv.x + hv.y * sv.y + hv.z * sv.z + hv.w * sv.w;
        d += hv.x * dv.x + hv.y * dv.y + hv.z * dv.z + hv.w * dv.w;
    }
    as_out[tid] = s;
    ad_out[tid] = d;
}

// ---- fills ----------------------------------------------------------
__global__ void fill_f32(float* p, float v, int n) {
    int t = blockIdx.x * blockDim.x + threadIdx.x;
    if (t < n) p[t] = v;
}
__global__ void fill_i32(int* p, int v, int n) {
    int t = blockIdx.x * blockDim.x + threadIdx.x;
    if (t < n) p[t] = v;
}

// ---- edge pass 1: leaky-relu logit + segment max (ordered-int) ------
__global__ void edge_pass1(const int* __restrict__ src,
                           const int* __restrict__ dst,
                           const float* __restrict__ as_v,
                           const float* __restrict__ ad_v,
                           float* __restrict__ ebuf,
                           int* __restrict__ mord)
{
    int tid = blockIdx.x * blockDim.x + threadIdx.x;
    if (tid >= ET * NH) return;
    int e = tid >> 2, h = tid & 3;
    int s = (e < NE) ? src[e] : (e - NE);
    int d = (e < NE) ? dst[e] : (e - NE);
    float v = as_v[s * NH + h] + ad_v[d * NH + h];
    v = (v > 0.f) ? v : 0.2f * v;              // leaky_relu(.,0.2)
    ebuf[tid] = v;
    atomicMax(&mord[d * NH + h], f2ord(v));
}

// ---- edge pass 2: exp(e - max) + segment sum ------------------------
__global__ void edge_pass2(const int* __restrict__ dst,
                           const int* __restrict__ mord,
                           float* __restrict__ ebuf,
                           float* __restrict__ denom)
{
    int tid = blockIdx.x * blockDim.x + threadIdx.x;
    if (tid >= ET * NH) return;
    int e = tid >> 2, h = tid & 3;
    int d = (e < NE) ? dst[e] : (e - NE);
    float m  = ord2f(mord[d * NH + h]);
    float ex = expf(ebuf[tid] - m);
    ebuf[tid] = ex;
    atomicAdd(&denom[d * NH + h], ex);
}

// ---- edge pass 3: alpha-scaled message scatter-add (float4 gather) ---
__global__ void edge_pass3(const int* __restrict__ src,
                           const int* __restrict__ dst,
                           const float* __restrict__ ebuf,
                           const float* __restrict__ denom,
                           const float* __restrict__ Hm,
                           float* __restrict__ agg)
{
    long long tid = (long long)blockIdx.x * blockDim.x + threadIdx.x;
    if (tid >= (long long)ET * (HID / 4)) return;
    int e = (int)(tid >> 5);                    // HID/4 = 32 quads per edge
    int q = (int)(tid & 31);
    int j = q << 2;                             // channel base (0..124)
    int h = j >> 5;                             // head
    int s = (e < NE) ? src[e] : (e - NE);
    int d = (e < NE) ? dst[e] : (e - NE);
    float alpha = ebuf[e * NH + h] / (denom[d * NH + h] + 1e-16f);
    float4 hv = *(const float4*)(Hm + (size_t)s * HID + j);   // global_load_b128
    float* ap = agg + (size_t)d * HID + j;
    atomicAdd(ap + 0, hv.x * alpha);
    atomicAdd(ap + 1, hv.y * alpha);
    atomicAdd(ap + 2, hv.z * alpha);
    atomicAdd(ap + 3, hv.w * alpha);
}

// ---- epilogue: bias + ELU -------------------------------------------
__global__ void bias_elu(const float* __restrict__ agg,
                         const float* __restrict__ b,
                         float* __restrict__ out)
{
    int tid = blockIdx.x * blockDim.x + threadIdx.x;
    if (tid >= NN * HID) return;
    int j = tid & 127;
    float y = agg[tid] + b[j];
    out[tid] = (y > 0.f) ? y : (expf(y) - 1.f);
}

// =====================================================================
static void run_gat_layer(const float* Xin, const float* W,
                          const float* a_s, const float* a_d, const float* b,
                          const int* src, const int* dst,
                          float* bufH, float* bufAgg, float* bufOut,
                          float* ebuf, float* asb, float* adb,
                          float* denom, int* mord, hipStream_t stream)
{
    const int TB = 256;
    // 1) projection GEMM (WMMA fp32 strip, W staged in LDS): bufH = Xin @ W
    int nMtiles = NN / 16;                                  // 3125 strip-waves
    int blocks  = (nMtiles * 32 + TB - 1) / TB;
    wmma_gemm_strip<HID, HID / 16><<<blocks, TB, HID * HID * sizeof(float), stream>>>(
        Xin, W, nullptr, bufH, nMtiles);
    // 2) attention logits
    alpha_kernel<<<(NN * NH + TB - 1) / TB, TB, 0, stream>>>(bufH, a_s, a_d, asb, adb);
    // 3) init reductions
    fill_i32<<<(NN * NH + TB - 1) / TB, TB, 0, stream>>>(mord, (int)0x80000000, NN * NH);
    fill_f32<<<(NN * NH + TB - 1) / TB, TB, 0, stream>>>(denom, 0.f, NN * NH);
    fill_f32<<<(NN * HID + TB - 1) / TB, TB, 0, stream>>>(bufAgg, 0.f, NN * HID);
    // 4) segment softmax + scatter
    int gEH = (ET * NH + TB - 1) / TB;
    edge_pass1<<<gEH, TB, 0, stream>>>(src, dst, asb, adb, ebuf, mord);
    edge_pass2<<<gEH, TB, 0, stream>>>(dst, mord, ebuf, denom);
    long long tot3 = (long long)ET * (HID / 4);
    edge_pass3<<<(unsigned)((tot3 + TB - 1) / TB), TB, 0, stream>>>(
        src, dst, ebuf, denom, bufH, bufAgg);
    // 5) bias + ELU
    bias_elu<<<(NN * HID + TB - 1) / TB, TB, 0, stream>>>(bufAgg, b, bufOut);
}

extern "C" void kernel_launch(void* const* d_in, const int* in_sizes, int n_in,
                              void* d_out, int out_size, void* d_ws, size_t ws_size,
                              hipStream_t stream)
{
    const float* x    = (const float*)d_in[0];
    const int*   edge = (const int*)  d_in[1];   // [2, E]
    const float* W1   = (const float*)d_in[2];
    const float* a_s1 = (const float*)d_in[3];
    const float* a_d1 = (const float*)d_in[4];
    const float* b1   = (const float*)d_in[5];
    const float* W2   = (const float*)d_in[6];
    const float* a_s2 = (const float*)d_in[7];
    const float* a_d2 = (const float*)d_in[8];
    const float* b2   = (const float*)d_in[9];
    const float* fc_w = (const float*)d_in[10];
    const float* fc_b = (const float*)d_in[11];
    float* out = (float*)d_out;

    const int* src = edge;
    const int* dst = edge + NE;

    // workspace carving (floats)
    float* bufH   = (float*)d_ws;            // NN*HID
    float* bufAgg = bufH   + (size_t)NN * HID;
    float* bufIn2 = bufAgg + (size_t)NN * HID;
    float* ebuf   = bufIn2 + (size_t)NN * HID;   // ET*NH
    float* asb    = ebuf   + (size_t)ET * NH;    // NN*NH
    float* adb    = asb    + (size_t)NN * NH;
    float* denom  = adb    + (size_t)NN * NH;
    int*   mord   = (int*)(denom + (size_t)NN * NH);

    // Layer 1: x -> bufIn2
    run_gat_layer(x, W1, a_s1, a_d1, b1, src, dst,
                  bufH, bufAgg, bufIn2, ebuf, asb, adb, denom, mord, stream);
    // Layer 2: bufIn2 -> bufIn2 (bufH/bufAgg reused as scratch)
    run_gat_layer(bufIn2, W2, a_s2, a_d2, b2, src, dst,
                  bufH, bufAgg, bufIn2, ebuf, asb, adb, denom, mord, stream);
    // Final FC (WMMA strip, fused bias): out = bufIn2 @ fc_w + fc_b  [NN x 32]
    const int TB = 256;
    int nMtiles = NN / 16;                                  // 3125
    int blocks  = (nMtiles * 32 + TB - 1) / TB;
    wmma_gemm_strip<HID, CH / 16><<<blocks, TB, HID * CH * sizeof(float), stream>>>(
        bufIn2, fc_w, fc_b, out, nMtiles);
}